// SGConv_PNY_21474836480038
// MI455X (gfx1250) — compile-verified
//
#include <hip/hip_runtime.h>

// ---------------- problem constants ----------------
#define NN      30000
#define DD      64
#define EE      480000
#define LL      16
#define TT      8
#define GG      (LL*TT)     // 128 groups
#define SPLITT  6

typedef __attribute__((ext_vector_type(2))) float v2f;
typedef __attribute__((ext_vector_type(8))) float v8f;

// D = A(16x4) * B(4x16) + C, all f32 (CDNA5 V_WMMA_F32_16X16X4_F32)
__device__ __forceinline__ v8f wmma4(v2f a, v2f b, v8f c) {
  return __builtin_amdgcn_wmma_f32_16x16x4_f32(false, a, false, b, (short)0, c,
                                               false, false);
}

// ---------------- 1. degree ----------------
__global__ void k_deg(const int* __restrict__ dst, float* __restrict__ deg) {
  int e = blockIdx.x * blockDim.x + threadIdx.x;
  if (e < EE) atomicAdd(&deg[dst[e]], 1.0f);
}

// ---------------- 2. norm + h ----------------
__global__ void k_norm_h(const float* __restrict__ feat,
                         const float* __restrict__ deg,
                         float* __restrict__ nrm, float* __restrict__ h) {
  int n = blockIdx.x, f = threadIdx.x;
  float nv = rsqrtf(fmaxf(deg[n], 1.0f));
  if (f == 0) nrm[n] = nv;
  h[n * DD + f] = feat[n * DD + f] * nv;
}

// ---------------- 3. agg[dst] += h[src] : 4 threads/edge ----------------
__global__ void k_scatter(const int* __restrict__ src, const int* __restrict__ dst,
                          const float* __restrict__ h, float* __restrict__ agg) {
  long long idx = (long long)blockIdx.x * blockDim.x + threadIdx.x;
  if (idx >= (long long)EE * 4) return;
  int e = (int)(idx >> 2), q = (int)(idx & 3);
  int s = src[e], d = dst[e];
  const float4* hp = (const float4*)(h + (size_t)s * DD + q * 16);
  float* base = agg + (size_t)d * DD + q * 16;
#pragma unroll
  for (int u = 0; u < 4; ++u) {
    float4 v = hp[u];
    atomicAdd(base + u * 4 + 0, v.x); atomicAdd(base + u * 4 + 1, v.y);
    atomicAdd(base + u * 4 + 2, v.z); atomicAdd(base + u * 4 + 3, v.w);
  }
}

// ---------------- 4. counting sort by g = label*T + time ----------------
__global__ void k_gcnt(const int* __restrict__ labels, const int* __restrict__ times,
                       int* __restrict__ cntg) {
  int n = blockIdx.x * blockDim.x + threadIdx.x;
  if (n < NN) atomicAdd(&cntg[labels[n] * TT + times[n]], 1);
}
__global__ void k_scan(const int* __restrict__ cntg, int* __restrict__ offs) {
  if (blockIdx.x == 0 && threadIdx.x == 0) {
    int acc = 0;
    for (int g = 0; g < GG; ++g) { offs[g] = acc; acc += cntg[g]; }
    offs[GG] = acc;
  }
}
__global__ void k_order(const int* __restrict__ labels, const int* __restrict__ times,
                        const int* __restrict__ offs, int* __restrict__ fill,
                        int* __restrict__ order) {
  int n = blockIdx.x * blockDim.x + threadIdx.x;
  if (n < NN) {
    int g = labels[n] * TT + times[n];
    order[offs[g] + atomicAdd(&fill[g], 1)] = n;
  }
}

// ---------------- 5. per-label feature sums ----------------
__global__ void k_sumY(const float* __restrict__ h, const int* __restrict__ offs,
                       const int* __restrict__ order, float* __restrict__ sumY) {
  int l = blockIdx.x, f = threadIdx.x;
  int s0 = offs[l * TT], s1 = offs[l * TT + TT];
  float s = 0.f;
  for (int n = s0; n < s1; ++n) s += h[(size_t)order[n] * DD + f];
  sumY[l * DD + f] = s;
}

// ---------------- 6. per-label SYRK S[l] = X_l^T X_l via WMMA ----------------
// one block/label; 16 waves = 4x4 tiles; 16-node stages via async-to-LDS.
#define HSTRD 72   // LDS row stride: 2*72 % 64 == 16 -> lane-halves hit disjoint banks
__global__ void __launch_bounds__(512) k_S(const float* __restrict__ h,
                                           const int* __restrict__ offs,
                                           const int* __restrict__ order,
                                           float* __restrict__ S) {
  __shared__ float hs[16][HSTRD];
  int l = blockIdx.x, tid = threadIdx.x;
  int wave = tid >> 5, lane = tid & 31;
  int i0 = (wave >> 2) * 16, j0 = (wave & 3) * 16;
  int s0 = offs[l * TT], s1 = offs[l * TT + TT];
  int kb = (lane < 16) ? 0 : 2, m = lane & 15;
  int e2 = tid * 2, srow = e2 >> 6, sf = e2 & 63;   // staging: 2 floats / thread
  v8f acc = {};
  int base = s0;
  for (; base + 16 <= s1; base += 16) {             // full stages: no divergence
    int node = order[base + srow];
    unsigned long long ga =
        (unsigned long long)(uintptr_t)(h + (size_t)node * DD + sf);
    unsigned lo = (unsigned)(uintptr_t)&hs[srow][sf];
    asm volatile("global_load_async_to_lds_b64 %0, %1, off"
                 :: "v"(lo), "v"(ga) : "memory");
    asm volatile("s_wait_asynccnt 0x0" ::: "memory");
    __syncthreads();
#pragma unroll
    for (int c = 0; c < 4; ++c) {
      int r0 = c * 4 + kb;
      v2f A, B;
      A.x = hs[r0][i0 + m];     A.y = hs[r0 + 1][i0 + m];
      B.x = hs[r0][j0 + m];     B.y = hs[r0 + 1][j0 + m];
      acc = wmma4(A, B, acc);
    }
    __syncthreads();
  }
  if (base < s1) {                                   // tail stage, zero-filled
    float v0 = 0.f, v1 = 0.f;
    int ia = base + srow;
    if (ia < s1) {
      int node = order[ia];
      v0 = h[(size_t)node * DD + sf];
      v1 = h[(size_t)node * DD + sf + 1];
    }
    hs[srow][sf] = v0; hs[srow][sf + 1] = v1;
    __syncthreads();
#pragma unroll
    for (int c = 0; c < 4; ++c) {
      int r0 = c * 4 + kb;
      v2f A, B;
      A.x = hs[r0][i0 + m];     A.y = hs[r0 + 1][i0 + m];
      B.x = hs[r0][j0 + m];     B.y = hs[r0 + 1][j0 + m];
      acc = wmma4(A, B, acc);
    }
  }
  for (int r = 0; r < 8; ++r) {
    int mi = (lane < 16) ? r : r + 8;
    S[(size_t)l * DD * DD + (i0 + mi) * DD + j0 + m] = acc[r];
  }
}

// ---------------- 7. prev_cov ----------------
__global__ void k_pcov(const float* __restrict__ S, const float* __restrict__ sumY,
                       const int* __restrict__ offs, float* __restrict__ pcov) {
  int idx = blockIdx.x * blockDim.x + threadIdx.x; // LL*DD*DD
  if (idx >= LL * DD * DD) return;
  int l = idx >> 12, ij = idx & 4095, i = ij >> 6, j = ij & 63;
  float cnt = (float)(offs[l * TT + TT] - offs[l * TT]);
  float mi = sumY[l * DD + i] / cnt, mj = sumY[l * DD + j] / cnt;
  pcov[idx] = (S[idx] - cnt * mi * mj) / (cnt - 1.0f);
}

// ---------------- 8. temp[a,t,b] = (P.w4)/(P.w2)^2 ----------------
__device__ __forceinline__ bool is_far(int t, int s) {
  int d = t - s; if (d < 0) d = -d;
  int lim = TT - 1 - t; if (t < lim) lim = t;
  return d > lim;
}
__global__ void k_temp(const float* __restrict__ P, float* __restrict__ temp) {
  int idx = blockIdx.x * blockDim.x + threadIdx.x; // LL*TT*LL = 2048
  if (idx >= LL * TT * LL) return;
  int a = idx >> 7, t = (idx >> 4) & 7, bI = idx & 15;
  float den = 0.f;
  for (int b2 = 0; b2 < LL; ++b2)
    for (int s = 0; s < TT; ++s)
      den += P[((a * TT + t) * LL + b2) * TT + s] * (is_far(t, s) ? 2.f : 1.f);
  float num = 0.f;
  for (int s = 0; s < TT; ++s)
    num += P[((a * TT + t) * LL + bI) * TT + s] * (is_far(t, s) ? 4.f : 1.f);
  temp[idx] = num / (den * den);
}

// ---------------- 9. cur_cov[a,t] = sum_b temp[a,t,b]*pcov[b] ----------------
__global__ void k_ccov(const float* __restrict__ temp, const float* __restrict__ pcov,
                       float* __restrict__ ccov) {
  int idx = blockIdx.x * blockDim.x + threadIdx.x; // GG*DD*DD
  if (idx >= GG * DD * DD) return;
  int lt = idx >> 12, ij = idx & 4095;
  float s = 0.f;
  for (int b = 0; b < LL; ++b)
    s += temp[lt * LL + b] * pcov[b * DD * DD + ij];
  ccov[idx] = s;
}

// ---------------- 10. batched 64x64 symmetric eigh: cyclic Jacobi ----------------
__global__ void __launch_bounds__(256) k_eigh(const float* __restrict__ ccov,
                                              float* __restrict__ lam,
                                              float* __restrict__ Qm) {
  __shared__ float Am[DD * DD];
  __shared__ float Vm[DD * DD];
  __shared__ int   top[32], bot[32], pp[32], qq[32];
  __shared__ float cc[32], ss[32];
  __shared__ float ev[DD];
  __shared__ int   prm[DD];
  int lt = blockIdx.x, tid = threadIdx.x;
  const float* A0 = ccov + (size_t)lt * DD * DD;
  for (int w = tid; w < DD * DD; w += 256) {
    Am[w] = A0[w];
    Vm[w] = ((w >> 6) == (w & 63)) ? 1.f : 0.f;
  }
  if (tid < 32) { top[tid] = 2 * tid; bot[tid] = 2 * tid + 1; }
  __syncthreads();
  for (int sweep = 0; sweep < 8; ++sweep) {
    for (int rnd = 0; rnd < 63; ++rnd) {
      if (tid < 32) {
        int p = top[tid], q = bot[tid];
        if (p > q) { int t = p; p = q; q = t; }
        pp[tid] = p; qq[tid] = q;
        float apq = Am[p * DD + q], c = 1.f, s = 0.f;
        if (fabsf(apq) > 1e-30f) {
          float tau = (Am[q * DD + q] - Am[p * DD + p]) / (2.f * apq);
          float tv = (tau >= 0.f ? 1.f : -1.f) / (fabsf(tau) + sqrtf(1.f + tau * tau));
          c = rsqrtf(1.f + tv * tv); s = tv * c;
        }
        cc[tid] = c; ss[tid] = s;
      }
      __syncthreads();
      for (int w = tid; w < 32 * DD; w += 256) {   // column rotations: A<-AG, V<-VG
        int k = w >> 6, i = w & 63;
        int p = pp[k], q = qq[k]; float c = cc[k], s = ss[k];
        float aip = Am[i * DD + p], aiq = Am[i * DD + q];
        Am[i * DD + p] = c * aip - s * aiq;
        Am[i * DD + q] = s * aip + c * aiq;
        float vip = Vm[i * DD + p], viq = Vm[i * DD + q];
        Vm[i * DD + p] = c * vip - s * viq;
        Vm[i * DD + q] = s * vip + c * viq;
      }
      __syncthreads();
      for (int w = tid; w < 32 * DD; w += 256) {   // row rotations: A<-G^T A
        int k = w >> 6, j = w & 63;
        int p = pp[k], q = qq[k]; float c = cc[k], s = ss[k];
        float apj = Am[p * DD + j], aqj = Am[q * DD + j];
        Am[p * DD + j] = c * apj - s * aqj;
        Am[q * DD + j] = s * apj + c * aqj;
      }
      __syncthreads();
      if (tid == 0) {                              // tournament rotation
        int nt[32], nb[32];
        nt[0] = top[0]; nt[1] = bot[0];
        for (int k = 2; k < 32; ++k) nt[k] = top[k - 1];
        for (int k = 0; k < 31; ++k) nb[k] = bot[k + 1];
        nb[31] = top[31];
        for (int k = 0; k < 32; ++k) { top[k] = nt[k]; bot[k] = nb[k]; }
      }
      __syncthreads();
    }
  }
  if (tid < DD) { ev[tid] = Am[tid * DD + tid]; prm[tid] = tid; }
  __syncthreads();
  if (tid == 0) {                                  // ascending sort (match eigh)
    for (int i = 1; i < DD; ++i) {
      float v = ev[i]; int pi = prm[i]; int j = i - 1;
      while (j >= 0 && ev[j] > v) { ev[j + 1] = ev[j]; prm[j + 1] = prm[j]; --j; }
      ev[j + 1] = v; prm[j + 1] = pi;
    }
  }
  __syncthreads();
  if (tid < DD) lam[(size_t)lt * DD + tid] = ev[tid];
  for (int w = tid; w < DD * DD; w += 256) {
    int i = w >> 6, k = w & 63;
    Qm[(size_t)lt * DD * DD + i * DD + k] = Vm[i * DD + prm[k]];
  }
}

// ---------------- 11. transform[l,t] = Q7 diag(sqrt(l7/lt)) Qt^T via WMMA ----------------
#define QSTRD 68   // row-gather pattern: 4*m distinct banks; halves disjoint parity
__global__ void __launch_bounds__(512) k_trf(const float* __restrict__ lam,
                                             const float* __restrict__ Qm,
                                             float* __restrict__ trf) {
  __shared__ float rsc[DD];
  __shared__ float q7s[DD][QSTRD];
  __shared__ float qts[DD][QSTRD];
  int lt = blockIdx.x, l = lt >> 3, tid = threadIdx.x;
  const float* Q7   = Qm  + ((size_t)l * TT + (TT - 1)) * DD * DD;
  const float* Qt   = Qm  + (size_t)lt * DD * DD;
  const float* lam7 = lam + ((size_t)l * TT + (TT - 1)) * DD;
  const float* lamt = lam + (size_t)lt * DD;
  if (tid < DD) rsc[tid] = sqrtf(lam7[tid] / lamt[tid]);
  __syncthreads();
  for (int w = tid; w < DD * DD; w += 512) {
    int i = w >> 6, k = w & 63;
    q7s[i][k] = Q7[w];
    qts[i][k] = Qt[w] * rsc[k];     // fold eigenvalue rescale into staged Qt
  }
  __syncthreads();
  int wave = tid >> 5, lane = tid & 31;
  int i0 = (wave >> 2) * 16, j0 = (wave & 3) * 16;
  int kb = (lane < 16) ? 0 : 2, m = lane & 15;
  v8f acc = {};
#pragma unroll
  for (int kc = 0; kc < 16; ++kc) {
    int k0 = kc * 4 + kb;
    v2f A, B;
    A.x = q7s[i0 + m][k0]; A.y = q7s[i0 + m][k0 + 1];
    B.x = qts[j0 + m][k0]; B.y = qts[j0 + m][k0 + 1];   // B[k][j] = Qt[j,k]*r[k]
    acc = wmma4(A, B, acc);
  }
  for (int r = 0; r < 8; ++r) {
    int mi = (lane < 16) ? r : r + 8;
    trf[(size_t)lt * DD * DD + (i0 + mi) * DD + j0 + m] = acc[r];
  }
}

// ---------------- 12. per-group mean of agg ----------------
__global__ void k_meantl(const float* __restrict__ agg, const int* __restrict__ offs,
                         const int* __restrict__ order, float* __restrict__ mtl) {
  int g = blockIdx.x, f = threadIdx.x;
  int s0 = offs[g], s1 = offs[g + 1];
  float s = 0.f;
  for (int n = s0; n < s1; ++n) s += agg[(size_t)order[n] * DD + f];
  float cnt = (float)(s1 - s0);
  mtl[g * DD + f] = s / fmaxf(cnt, 1.0f);
}

// ---------------- 13. grouped whitening via WMMA: h2 = sel(M(x-mu)+mu, x)*norm ----------------
__global__ void __launch_bounds__(128) k_apply(const float* __restrict__ agg,
                                               const float* __restrict__ trf,
                                               const float* __restrict__ mtl,
                                               const int* __restrict__ offs,
                                               const int* __restrict__ order,
                                               const int* __restrict__ times,
                                               const float* __restrict__ nrm,
                                               float* __restrict__ h2) {
  __shared__ float mu_s[DD];
  __shared__ float xs[16][QSTRD];      // (x - mu) tile, node-major
  int g = blockIdx.x, tid = threadIdx.x;
  if (tid < DD) mu_s[tid] = mtl[g * DD + tid];
  __syncthreads();
  int wave = tid >> 5, lane = tid & 31;
  int i0 = wave * 16;                  // output feature tile
  int kb = (lane < 16) ? 0 : 2, m = lane & 15;
  const float* M = trf + (size_t)g * DD * DD;
  v2f bf[16];
#pragma unroll
  for (int kc = 0; kc < 16; ++kc) {    // B[k][i] = M[i][k]
    float2 w = *(const float2*)(M + (i0 + m) * DD + kc * 4 + kb);
    bf[kc].x = w.x; bf[kc].y = w.y;
  }
  int s0 = offs[g], s1 = offs[g + 1];
  for (int base = s0; base < s1; base += 16) {
    for (int w = tid; w < 16 * DD; w += 128) {     // cooperative tile stage
      int row = w >> 6, f = w & 63;
      int ia = base + row;
      float v = 0.f;
      if (ia < s1) v = agg[(size_t)order[ia] * DD + f] - mu_s[f];
      xs[row][f] = v;
    }
    __syncthreads();
    v8f acc = {};
#pragma unroll
    for (int kc = 0; kc < 16; ++kc) {
      int k0 = kc * 4 + kb;
      v2f A;
      A.x = xs[m][k0]; A.y = xs[m][k0 + 1];
      acc = wmma4(A, bf[kc], acc);
    }
    for (int r = 0; r < 8; ++r) {
      int mi = (lane < 16) ? r : r + 8;
      int idx = base + mi;
      if (idx < s1) {
        int n = order[idx], col = i0 + m;
        float v  = acc[r] + mu_s[col];
        float xv = agg[(size_t)n * DD + col];
        float sel = (times[n] < SPLITT) ? v : xv;
        h2[(size_t)n * DD + col] = sel * nrm[n];
      }
    }
    __syncthreads();
  }
}

// ---------------- 14. column stats ----------------
__global__ void k_colstats(const float* __restrict__ h2, float* __restrict__ csum,
                           float* __restrict__ csq) {
  int f = threadIdx.x;
  int n0 = blockIdx.x * 256, n1 = min(n0 + 256, NN);
  float s = 0.f, s2 = 0.f;
  for (int n = n0; n < n1; ++n) {
    float v = h2[(size_t)n * DD + f];
    s += v; s2 += v * v;
  }
  atomicAdd(&csum[f], s);
  atomicAdd(&csq[f], s2);
}
__global__ void k_colfin(const float* __restrict__ csum, const float* __restrict__ csq,
                         float* __restrict__ cmean, float* __restrict__ cistd) {
  int f = threadIdx.x;
  float mean = csum[f] / (float)NN;
  float var = (csq[f] - (float)NN * mean * mean) / ((float)NN - 1.0f);
  cmean[f] = mean;
  cistd[f] = rsqrtf(var);
}

// ---------------- 15. out = z @ W^T + b via WMMA ----------------
__global__ void __launch_bounds__(128) k_out(const float* __restrict__ h2,
                                             const float* __restrict__ W,
                                             const float* __restrict__ bias,
                                             const float* __restrict__ cmean,
                                             const float* __restrict__ cistd,
                                             float* __restrict__ out) {
  int tile = blockIdx.x;                 // 16-node tile
  int wave = threadIdx.x >> 5, lane = threadIdx.x & 31;
  int o0 = wave * 16;                    // output column tile
  int kb = (lane < 16) ? 0 : 2, m = lane & 15;
  v2f wf[16];
#pragma unroll
  for (int kc = 0; kc < 16; ++kc) {      // B[k][o] = W[o][k]
    float2 w = *(const float2*)(W + (o0 + m) * DD + kc * 4 + kb);
    wf[kc].x = w.x; wf[kc].y = w.y;
  }
  int n = tile * 16 + m;
  bool ok = (n < NN);
  v8f acc = {};
#pragma unroll
  for (int kc = 0; kc < 16; ++kc) {
    int k0 = kc * 4 + kb;
    v2f A; A.x = 0.f; A.y = 0.f;
    if (ok) {
      float2 z = *(const float2*)(h2 + (size_t)n * DD + k0);
      A.x = (z.x - cmean[k0])     * cistd[k0];
      A.y = (z.y - cmean[k0 + 1]) * cistd[k0 + 1];
    }
    acc = wmma4(A, wf[kc], acc);
  }
  for (int r = 0; r < 8; ++r) {
    int mi = (lane < 16) ? r : r + 8;
    int nn = tile * 16 + mi;
    if (nn < NN) out[(size_t)nn * DD + o0 + m] = acc[r] + bias[o0 + m];
  }
}

// ---------------- launch ----------------
extern "C" void kernel_launch(void* const* d_in, const int* in_sizes, int n_in,
                              void* d_out, int out_size, void* d_ws, size_t ws_size,
                              hipStream_t stream) {
  const float* feat  = (const float*)d_in[0];
  const float* W     = (const float*)d_in[1];
  const float* bias  = (const float*)d_in[2];
  const float* P     = (const float*)d_in[3];
  const int*   src   = (const int*)d_in[4];
  const int*   dst   = (const int*)d_in[5];
  const int*   labels= (const int*)d_in[6];
  const int*   times = (const int*)d_in[7];
  float* out = (float*)d_out;

  float* ws   = (float*)d_ws;
  float* deg  = ws;                 // N
  float* nrm  = deg  + NN;          // N
  float* h    = nrm  + NN;          // N*D
  float* agg  = h    + (size_t)NN * DD;
  float* sumY = agg  + (size_t)NN * DD;        // L*D
  float* S    = sumY + LL * DD;                // L*D*D
  float* pcov = S    + LL * DD * DD;           // L*D*D
  float* temp = pcov + LL * DD * DD;           // L*T*L
  float* ccov = temp + LL * TT * LL;           // G*D*D
  float* lam  = ccov + (size_t)GG * DD * DD;   // G*D
  float* Qm   = lam  + GG * DD;                // G*D*D
  float* trf  = Qm   + (size_t)GG * DD * DD;   // G*D*D
  float* mtl  = trf  + (size_t)GG * DD * DD;   // G*D
  float* h2   = mtl  + GG * DD;                // N*D
  float* csum = h2   + (size_t)NN * DD;        // D
  float* csq  = csum + DD;                     // D
  float* cmean= csq  + DD;                     // D
  float* cistd= cmean+ DD;                     // D
  int* cntg  = (int*)(cistd + DD);             // G
  int* offs  = cntg + GG;                      // G+1
  int* fill  = offs + GG + 1;                  // G
  int* order = fill + GG;                      // N

  hipMemsetAsync(deg,  0, sizeof(float) * NN, stream);
  hipMemsetAsync(agg,  0, sizeof(float) * (size_t)NN * DD, stream);
  hipMemsetAsync(csum, 0, sizeof(float) * 2 * DD, stream);   // csum+csq contiguous
  hipMemsetAsync(cntg, 0, sizeof(int) * GG, stream);
  hipMemsetAsync(fill, 0, sizeof(int) * GG, stream);

  k_deg      <<<(EE + 255) / 256, 256, 0, stream>>>(dst, deg);
  k_norm_h   <<<NN, DD, 0, stream>>>(feat, deg, nrm, h);
  k_scatter  <<<(EE * 4 + 255) / 256, 256, 0, stream>>>(src, dst, h, agg);
  k_gcnt     <<<(NN + 255) / 256, 256, 0, stream>>>(labels, times, cntg);
  k_scan     <<<1, 1, 0, stream>>>(cntg, offs);
  k_order    <<<(NN + 255) / 256, 256, 0, stream>>>(labels, times, offs, fill, order);
  k_sumY     <<<LL, DD, 0, stream>>>(h, offs, order, sumY);
  k_S        <<<LL, 512, 0, stream>>>(h, offs, order, S);
  k_pcov     <<<(LL * DD * DD + 255) / 256, 256, 0, stream>>>(S, sumY, offs, pcov);
  k_temp     <<<(LL * TT * LL + 255) / 256, 256, 0, stream>>>(P, temp);
  k_ccov     <<<(GG * DD * DD + 255) / 256, 256, 0, stream>>>(temp, pcov, ccov);
  k_eigh     <<<GG, 256, 0, stream>>>(ccov, lam, Qm);
  k_trf      <<<GG, 512, 0, stream>>>(lam, Qm, trf);
  k_meantl   <<<GG, DD, 0, stream>>>(agg, offs, order, mtl);
  k_apply    <<<GG, 128, 0, stream>>>(agg, trf, mtl, offs, order, times, nrm, h2);
  k_colstats <<<(NN + 255) / 256, DD, 0, stream>>>(h2, csum, csq);
  k_colfin   <<<1, DD, 0, stream>>>(csum, csq, cmean, cistd);
  k_out      <<<(NN + 15) / 16, 128, 0, stream>>>(h2, W, bias, cmean, cistd, out);
}